// ReverseConv_46643344835240
// MI455X (gfx1250) — compile-verified
//
#include <hip/hip_runtime.h>

// ReverseConv via CGNR (conjugate gradient on the normal equations) for the
// 16ch x 16x16, 3x3, pad-1 conv operator A:  solve (A^T A) y = A^T (x - bias).
// A and A^T applied as implicit GEMMs with V_WMMA_F32_16X16X4_F32 (fp32 path).
//
// This revision removes all runtime im2col index decode:
//  * K padded to 12 slots per input channel (3 di x 4 dj; dj==3 slot = 0 weight)
//    -> ci, di are compile-time constants in the unrolled K loop (12 % 4 == 0),
//       only dj = 2*half + t is lane-dependent and folds into the base address.
//  * conv inputs held in a zero-bordered [16][18][18] LDS layout
//    -> gathers are pure constant-offset ds_loads, no cndmask padding checks.
// All solver state LDS-resident in one WGP; single launch; deterministic.

typedef __attribute__((ext_vector_type(2))) float v2f;
typedef __attribute__((ext_vector_type(8))) float v8f;

#define NT    512            // 16 wave32 waves: one wave per output row
#define NITER 384            // fixed CG iteration count (deterministic)
#define CPAD  324            // padded channel stride = 18*18
#define IIMG  (16 * CPAD)    // padded image floats (borders zero)
#define IPAD  (IIMG + 32)    // + slack so dj==3 junk reads stay in-allocation
#define WK    192            // padded K = 16 ci * 12

// Padded-image index of flat element v (c,y,x in 16x16x16).
__device__ __forceinline__ int pidx(int v) {
    return ((v >> 8) * CPAD) + (((v >> 4) & 15) * 18) + (v & 15) + 19; // (+1,+1) border
}

// Out[co][i][j] = sum_k Wm[co][k] * In_padded[ci][i+di][l+dj],
// k = ci*12 + di*4 + dj.  Wave `wave` computes output row i as a
// 16(co) x 16(j) tile in 48 WMMA steps of K=4.
__device__ __forceinline__ void conv_wmma(const float* __restrict__ Wm,  // LDS [16][192]
                                          const float* __restrict__ In,  // LDS [16][18][18]
                                          float*       __restrict__ Out, // LDS
                                          int oStrideC, int oStrideY, int oOff,
                                          int lane, int wave)
{
    const int half = lane >> 4;
    const int l    = lane & 15;
    const int i    = wave;

    // Loop-invariant bases; per-step addresses are base + compile-time offset.
    const float* wRow  = Wm + (l * WK + 2 * half);        // A: M=l, K=4kk+2*half(+1)
    const float* inRow = In + (i * 18 + l + 2 * half);    // B: N=l, dj=2*half(+1)

    v8f acc = {};
    #pragma unroll
    for (int kk = 0; kk < 48; ++kk) {
        const int ci = kk / 3;        // compile-time after unroll
        const int di = kk - ci * 3;   // compile-time after unroll
        v2f a, b;
        a.x = wRow[kk * 4];
        a.y = wRow[kk * 4 + 1];
        const int ioff = ci * CPAD + di * 18;  // compile-time immediate
        b.x = inRow[ioff];
        b.y = inRow[ioff + 1];
        acc = __builtin_amdgcn_wmma_f32_16x16x4_f32(
                  false, a, false, b, (short)0, acc, false, false);
    }
    // C/D layout: VGPR v -> M = v (lanes 0-15) / v+8 (lanes 16-31), N = l.
    #pragma unroll
    for (int v = 0; v < 8; ++v)
        Out[(v + 8 * half) * oStrideC + (i + oOff) * oStrideY + (l + oOff)] = acc[v];
}

__device__ __forceinline__ float block_sum(float v, float* red, int tid)
{
    red[tid] = v;
    __syncthreads();
    #pragma unroll
    for (int off = NT / 2; off >= 1; off >>= 1) {
        if (tid < off) red[tid] += red[tid + off];
        __syncthreads();
    }
    const float s = red[0];
    __syncthreads();
    return s;
}

__global__ __launch_bounds__(NT)
void ReverseConv_46643344835240_kernel(const float* __restrict__ x,
                                       const float* __restrict__ inp,
                                       const float* __restrict__ weight,
                                       const float* __restrict__ bias,
                                       float* __restrict__ out)
{
    __shared__ float Wm[16 * WK];    // forward:  Wm[co][ci*12+di*4+dj]
    __shared__ float Wt[16 * WK];    // adjoint:  Wt[ci][co*12+(2-di)*4+(2-dj)]
    __shared__ float Pp[IPAD];       // search direction, padded image layout
    __shared__ float Qp[IPAD];       // q = A p (and initially b), padded layout
    __shared__ float Y [4096];       // solution
    __shared__ float Rv[4096];       // residual
    __shared__ float Sv[4096];       // s = A^T q
    __shared__ float red[NT];

    const int tid  = threadIdx.x;
    const int lane = tid & 31;
    const int wave = tid >> 5;

    // Zero padded buffers (borders/slack stay zero forever) and weight pads.
    for (int v = tid; v < IPAD; v += NT) { Pp[v] = 0.0f; Qp[v] = 0.0f; }
    for (int v = tid; v < 16 * WK; v += NT) { Wm[v] = 0.0f; Wt[v] = 0.0f; }
    __syncthreads();

    // Scatter weights into padded layouts (unique slots per thread -> no race).
    for (int idx = tid; idx < 2304; idx += NT) {
        const int co  = idx / 144;
        const int k   = idx - co * 144;
        const int ci  = k / 9;
        const int rem = k - ci * 9;
        const int di  = rem / 3;
        const int dj  = rem - di * 3;
        const float wv = weight[idx];                    // [co][ci][di][dj] flat
        Wm[co * WK + ci * 12 + di * 4 + dj] = wv;
        Wt[ci * WK + co * 12 + (2 - di) * 4 + (2 - dj)] = wv;
    }
    // b = x - bias into Qp interior; y = 0.
    for (int v = tid; v < 4096; v += NT) {
        Qp[pidx(v)] = x[v] - bias[v >> 8];
        Y[v] = 0.0f;
    }
    __syncthreads();

    // r0 = A^T b ; p0 = r0.
    conv_wmma(Wt, Qp, Rv, 256, 16, 0, lane, wave);
    __syncthreads();
    for (int v = tid; v < 4096; v += NT) Pp[pidx(v)] = Rv[v];
    __syncthreads();

    float rr;
    {
        float s = 0.0f;
        #pragma unroll
        for (int v = tid; v < 4096; v += NT) s += Rv[v] * Rv[v];
        rr = block_sum(s, red, tid);
    }

    #pragma unroll 1
    for (int it = 0; it < NITER; ++it) {
        conv_wmma(Wm, Pp, Qp, CPAD, 18, 1, lane, wave);   // q = A p (padded out)
        __syncthreads();

        float sq = 0.0f;                                   // qq = ||A p||^2
        for (int v = tid; v < IIMG; v += NT) sq += Qp[v] * Qp[v]; // borders are 0
        const float qq = block_sum(sq, red, tid);

        conv_wmma(Wt, Qp, Sv, 256, 16, 0, lane, wave);     // s = A^T A p
        __syncthreads();

        const float alpha = (qq > 0.0f) ? (rr / qq) : 0.0f;
        for (int v = tid; v < 4096; v += NT) {
            Y[v]  += alpha * Pp[pidx(v)];
            Rv[v] -= alpha * Sv[v];
        }
        __syncthreads();

        float sr = 0.0f;
        #pragma unroll
        for (int v = tid; v < 4096; v += NT) sr += Rv[v] * Rv[v];
        const float rr_new = block_sum(sr, red, tid);
        const float beta   = (rr > 0.0f) ? (rr_new / rr) : 0.0f;

        for (int v = tid; v < 4096; v += NT) {
            const int p = pidx(v);
            Pp[p] = Rv[v] + beta * Pp[p];
        }
        __syncthreads();
        rr = rr_new;
    }

    // Reference fallback: where(max|y| == 0, y + inp, y).
    float m = 0.0f;
    for (int v = tid; v < 4096; v += NT) m = fmaxf(m, fabsf(Y[v]));
    red[tid] = m;
    __syncthreads();
    #pragma unroll
    for (int off = NT / 2; off >= 1; off >>= 1) {
        if (tid < off) red[tid] = fmaxf(red[tid], red[tid + off]);
        __syncthreads();
    }
    const float mx = red[0];
    __syncthreads();

    for (int v = tid; v < 4096; v += NT)
        out[v] = (mx == 0.0f) ? (Y[v] + inp[v]) : Y[v];
}

extern "C" void kernel_launch(void* const* d_in, const int* in_sizes, int n_in,
                              void* d_out, int out_size, void* d_ws, size_t ws_size,
                              hipStream_t stream) {
    const float* x      = (const float*)d_in[0];   // (1,16,16,16)
    const float* inp    = (const float*)d_in[1];   // (1,16,16,16)
    const float* weight = (const float*)d_in[2];   // (16,16,3,3)
    const float* bias   = (const float*)d_in[3];   // (16,)
    float* out = (float*)d_out;                    // 4096 f32
    (void)in_sizes; (void)n_in; (void)out_size; (void)d_ws; (void)ws_size;

    ReverseConv_46643344835240_kernel<<<1, NT, 0, stream>>>(x, inp, weight, bias, out);
}